// SimpleGCN_16063177687398
// MI455X (gfx1250) — compile-verified
//
#include <hip/hip_runtime.h>

typedef __attribute__((ext_vector_type(2))) float v2f;
typedef __attribute__((ext_vector_type(8))) float v8f;

// ---------------------------------------------------------------- zero
__global__ void gcn_zero(float* __restrict__ p, long long n) {
    long long i = (long long)blockIdx.x * blockDim.x + threadIdx.x;
    long long stride = (long long)gridDim.x * blockDim.x;
    for (; i < n; i += stride) p[i] = 0.0f;
}

// ---------------------------------------------------------------- degree (edges + self loops)
__global__ void gcn_degree(const int* __restrict__ dst, long long E, int N,
                           float* __restrict__ deg) {
    long long total = E + N;
    long long i = (long long)blockIdx.x * blockDim.x + threadIdx.x;
    long long stride = (long long)gridDim.x * blockDim.x;
    for (; i < total; i += stride) {
        int d = (i < E) ? dst[i] : (int)(i - E);
        atomicAdd(deg + d, 1.0f);
    }
}

// ---------------------------------------------------------------- deg -> rsqrt(deg) in place
__global__ void gcn_rsqrt(float* __restrict__ deg, int N) {
    int i = blockIdx.x * blockDim.x + threadIdx.x;
    if (i < N) {
        float d = deg[i];
        deg[i] = (d > 0.0f) ? rsqrtf(d) : 0.0f;
    }
}

// ---------------------------------------------------------------- GEMM1: h1[N,16] = x[N,128] @ W1[128,16]
// One wave per 16-row tile; K=128 via 32 x V_WMMA_F32_16X16X4_F32.
__global__ void gcn_gemm1_wmma(const float* __restrict__ x, const float* __restrict__ W1,
                               float* __restrict__ h1, int N, int ntiles) {
    int wave = (int)(((long long)blockIdx.x * blockDim.x + threadIdx.x) >> 5);
    if (wave >= ntiles) return;
    int lane = threadIdx.x & 31;
    int l15  = lane & 15;
    int half = lane >> 4;              // 0: K pair {0,1}, 1: K pair {2,3}
    int row  = wave * 16 + l15;        // A-matrix row for this lane
    int rowc = (row < N) ? row : (N - 1);
    const float* xrow = x + (long long)rowc * 128;

    v8f c = {0.f, 0.f, 0.f, 0.f, 0.f, 0.f, 0.f, 0.f};
#pragma unroll
    for (int k0 = 0; k0 < 128; k0 += 4) {
        int ka = k0 + 2 * half;
        v2f a = *(const v2f*)(xrow + ka);          // A[m][ka], A[m][ka+1]
        v2f b;
        b.x = W1[ka * 16 + l15];                   // B[ka][n]
        b.y = W1[(ka + 1) * 16 + l15];             // B[ka+1][n]
        c = __builtin_amdgcn_wmma_f32_16x16x4_f32(false, a, false, b, (short)0, c,
                                                  false, false);
    }
#pragma unroll
    for (int v = 0; v < 8; ++v) {
        int m = v + 8 * half;                      // D row within tile
        int r = wave * 16 + m;
        if (r < N) h1[(long long)r * 16 + l15] = c[v];
    }
}

// ---------------------------------------------------------------- edge scatter: acc[dst] += coef * h[src]
__global__ void gcn_aggregate(const float* __restrict__ h, const int* __restrict__ src,
                              const int* __restrict__ dst, const float* __restrict__ dinv,
                              float* __restrict__ acc, long long E) {
    long long i = (long long)blockIdx.x * blockDim.x + threadIdx.x;
    long long stride = (long long)gridDim.x * blockDim.x;
    for (; i < E; i += stride) {
        int s = src[i];
        int d = dst[i];
        float cf = dinv[s] * dinv[d];
        const float4* hs = (const float4*)(h + (long long)s * 16);
        float4 a0 = hs[0], a1 = hs[1], a2 = hs[2], a3 = hs[3];
        float* o = acc + (long long)d * 16;
        atomicAdd(o + 0,  cf * a0.x); atomicAdd(o + 1,  cf * a0.y);
        atomicAdd(o + 2,  cf * a0.z); atomicAdd(o + 3,  cf * a0.w);
        atomicAdd(o + 4,  cf * a1.x); atomicAdd(o + 5,  cf * a1.y);
        atomicAdd(o + 6,  cf * a1.z); atomicAdd(o + 7,  cf * a1.w);
        atomicAdd(o + 8,  cf * a2.x); atomicAdd(o + 9,  cf * a2.y);
        atomicAdd(o + 10, cf * a2.z); atomicAdd(o + 11, cf * a2.w);
        atomicAdd(o + 12, cf * a3.x); atomicAdd(o + 13, cf * a3.y);
        atomicAdd(o + 14, cf * a3.z); atomicAdd(o + 15, cf * a3.w);
    }
}

// ---------------------------------------------------------------- finalize1: r = relu(agg1 + h1*dinv^2 + b1)   (in place on agg1)
__global__ void gcn_finalize1(const float* __restrict__ h1, const float* __restrict__ dinv,
                              const float* __restrict__ b1, float* __restrict__ r,
                              long long n16) {
    long long i = (long long)blockIdx.x * blockDim.x + threadIdx.x;
    if (i >= n16) return;
    long long node = i >> 4;
    int k = (int)(i & 15);
    float di = dinv[node];
    float v = r[i] + h1[i] * di * di + b1[k];
    r[i] = (v > 0.0f) ? v : 0.0f;
}

// ---------------------------------------------------------------- finalize2: z = agg2 + r*dinv^2   (in place on agg2)
__global__ void gcn_finalize2(const float* __restrict__ r, const float* __restrict__ dinv,
                              float* __restrict__ z, long long n16) {
    long long i = (long long)blockIdx.x * blockDim.x + threadIdx.x;
    if (i >= n16) return;
    long long node = i >> 4;
    float di = dinv[node];
    z[i] = z[i] + r[i] * di * di;
}

// ---------------------------------------------------------------- GEMM2: out[N,C] = z[N,16] @ W2[16,C] + b2
// One wave per (16-row, 16-col) tile; K=16 via 4 WMMA steps; C=138 -> 9 col tiles.
__global__ void gcn_gemm2_wmma(const float* __restrict__ z, const float* __restrict__ W2,
                               const float* __restrict__ b2, float* __restrict__ out,
                               int N, int ntiles, int C, int ctiles) {
    int wave = (int)(((long long)blockIdx.x * blockDim.x + threadIdx.x) >> 5);
    if (wave >= ntiles * ctiles) return;
    int it = wave / ctiles;
    int jt = wave % ctiles;
    int lane = threadIdx.x & 31;
    int l15  = lane & 15;
    int half = lane >> 4;
    int row  = it * 16 + l15;
    int rowc = (row < N) ? row : (N - 1);
    int col  = jt * 16 + l15;
    const float* zrow = z + (long long)rowc * 16;

    v8f c = {0.f, 0.f, 0.f, 0.f, 0.f, 0.f, 0.f, 0.f};
#pragma unroll
    for (int k0 = 0; k0 < 16; k0 += 4) {
        int ka = k0 + 2 * half;
        v2f a = *(const v2f*)(zrow + ka);
        v2f b;
        b.x = (col < C) ? W2[ka * C + col] : 0.0f;
        b.y = (col < C) ? W2[(ka + 1) * C + col] : 0.0f;
        c = __builtin_amdgcn_wmma_f32_16x16x4_f32(false, a, false, b, (short)0, c,
                                                  false, false);
    }
    float bias = (col < C) ? b2[col] : 0.0f;
#pragma unroll
    for (int v = 0; v < 8; ++v) {
        int m = v + 8 * half;
        int r = it * 16 + m;
        if (r < N && col < C) out[(long long)r * C + col] = c[v] + bias;
    }
}

// ================================================================ launch
extern "C" void kernel_launch(void* const* d_in, const int* in_sizes, int n_in,
                              void* d_out, int out_size, void* d_ws, size_t ws_size,
                              hipStream_t stream) {
    const float* x  = (const float*)d_in[0];
    const int*   ei = (const int*)d_in[1];     // [2, E] int32
    const float* W1 = (const float*)d_in[2];
    const float* b1 = (const float*)d_in[3];
    const float* W2 = (const float*)d_in[4];
    const float* b2 = (const float*)d_in[5];

    const int H = in_sizes[3];                 // 16
    const int F = in_sizes[2] / H;             // 128
    const int C = in_sizes[5];                 // 138
    const int N = in_sizes[0] / F;             // 100000
    const long long E = (long long)in_sizes[1] / 2;

    const int* srcA = ei;
    const int* dstA = ei + E;

    float* ws  = (float*)d_ws;
    float* deg = ws;                           // [N]      -> becomes dinv
    float* h1  = ws + N;                       // [N,16]
    float* r   = ws + 17LL * N;                // [N,16]   agg1 -> relu output
    float* z   = ws + 33LL * N;                // [N,16]   agg2 -> pre-GEMM2

    const int ntiles = (N + 15) / 16;          // 6250
    const int ctiles = (C + 15) / 16;          // 9
    const int B = 256;

    // 1) zero deg + h1 + agg1 + agg2 regions (49N floats)
    {
        long long zn = 49LL * N;
        int blocks = (int)((zn + B - 1) / B);
        gcn_zero<<<blocks, B, 0, stream>>>(ws, zn);
    }
    // 2) degrees (edges + self loops)
    {
        long long total = E + N;
        int blocks = (int)((total + B - 1) / B);
        gcn_degree<<<blocks, B, 0, stream>>>(dstA, E, N, deg);
    }
    // 3) dinv = rsqrt(deg)
    gcn_rsqrt<<<(N + B - 1) / B, B, 0, stream>>>(deg, N);
    // 4) h1 = x @ W1  (WMMA)
    {
        long long threads = (long long)ntiles * 32;
        gcn_gemm1_wmma<<<(int)((threads + B - 1) / B), B, 0, stream>>>(x, W1, h1, N, ntiles);
    }
    // 5) agg1 += coef * h1[src]
    gcn_aggregate<<<(int)((E + B - 1) / B), B, 0, stream>>>(h1, srcA, dstA, deg, r, E);
    // 6) r = relu(agg1 + self-loop + b1)
    {
        long long n16 = 16LL * N;
        gcn_finalize1<<<(int)((n16 + B - 1) / B), B, 0, stream>>>(h1, deg, b1, r, n16);
    }
    // 7) agg2 += coef * r[src]   (aggregate BEFORE W2: linearity)
    gcn_aggregate<<<(int)((E + B - 1) / B), B, 0, stream>>>(r, srcA, dstA, deg, z, E);
    // 8) z = agg2 + self-loop
    {
        long long n16 = 16LL * N;
        gcn_finalize2<<<(int)((n16 + B - 1) / B), B, 0, stream>>>(r, deg, z, n16);
    }
    // 9) out = z @ W2 + b2  (WMMA)
    {
        long long threads = (long long)ntiles * ctiles * 32;
        gcn_gemm2_wmma<<<(int)((threads + B - 1) / B), B, 0, stream>>>(
            z, W2, b2, (float*)d_out, N, ntiles, C, ctiles);
    }
}